// S4DBLK_27797028340352
// MI455X (gfx1250) — compile-verified
//
#include <hip/hip_runtime.h>
#include <cmath>

#define B_   8
#define H_   256
#define L_   8192
#define N2_  32
#define NSEG 4
#define LSEG (L_ / NSEG)          // 2048
#define NCOL 64                   // stage-2 column tile
#define S2   72                   // LDS row stride (2*S2 mod 64 == 16 -> bank-safe halves)
#define LN_EPS 1e-5f

typedef __attribute__((ext_vector_type(2))) float v2f;
typedef __attribute__((ext_vector_type(8))) float v8f;

// ---- CDNA5 async global->LDS path (guarded, plain-load fallback) ----
#if defined(__has_builtin)
#if __has_builtin(__builtin_amdgcn_global_load_async_to_lds_b32) && \
    __has_builtin(__builtin_amdgcn_s_wait_asynccnt)
#define HAVE_ASYNC_LDS 1
#endif
#endif
#ifndef HAVE_ASYNC_LDS
#define HAVE_ASYNC_LDS 0
#endif

#if HAVE_ASYNC_LDS
typedef __attribute__((address_space(1))) int g_int;   // printed as '__device__ int *'
typedef __attribute__((address_space(3))) int l_int;   // printed as '__shared__ int *'
__device__ __forceinline__ void async_copy_f32(const float* gsrc, float* ldst) {
    __builtin_amdgcn_global_load_async_to_lds_b32(
        (g_int*)(void*)gsrc,
        (l_int*)(void*)ldst,
        0, 0);
}
#endif

// ---------------------------------------------------------------------------
// Per-mode S4D constants (shared by stage-1 kernels).
// w = exp(dt*A), C2 = 2*C*expm1(dt*A)/A  (recurrence == FFT causal conv)
// ---------------------------------------------------------------------------
struct ModeK { float wr, wi, dAr, dAi; };

__device__ __forceinline__ ModeK mode_setup(int h, int lane,
    const float* log_dt, const float* log_A_real, const float* A_imag) {
    ModeK m;
    const int   hn  = h * N2_ + lane;
    const float dt  = expf(log_dt[h]);
    const float Ar  = -expf(log_A_real[hn]);
    const float Ai  = A_imag[hn];
    m.dAr = Ar * dt;
    m.dAi = Ai * dt;
    const float ew = expf(m.dAr);
    m.wr = ew * cosf(m.dAi);
    m.wi = ew * sinf(m.dAi);
    return m;
}

// ---------------------------------------------------------------------------
// Stage 1a: state-only pass. One wave per (b,h,seg) for seg=0..2; scans its
// 2048-step segment with zero initial state, records end-state per mode.
// ---------------------------------------------------------------------------
__global__ __launch_bounds__(256) void s4d_state_kernel(
    const float* __restrict__ u,
    const float* __restrict__ log_dt,
    const float* __restrict__ log_A_real,
    const float* __restrict__ A_imag,
    float2* __restrict__ Sloc)             // (B*H, 3, 32) complex end-states
{
    const int lane = threadIdx.x & 31;
    const int w    = threadIdx.x >> 5;
    const int wg   = blockIdx.x * 8 + w;   // [0, B*H*3)
    const int seg  = wg % 3;
    const int pair = wg / 3;               // b*H + h
    const int h    = pair & (H_ - 1);

    const ModeK mk = mode_setup(h, lane, log_dt, log_A_real, A_imag);

    float sr = 0.0f, si = 0.0f;
    const int base = pair * L_ + seg * LSEG;
    for (int lc = 0; lc < LSEG; lc += 32) {
        const float uvec = u[base + lc + lane];
#pragma unroll
        for (int j = 0; j < 32; ++j) {
            const float uv  = __shfl(uvec, j, 32);
            const float t_r = fmaf(mk.wr, sr, fmaf(-mk.wi, si, uv));
            const float t_i = fmaf(mk.wr, si, mk.wi * sr);
            sr = t_r; si = t_i;
        }
    }
    Sloc[(pair * 3 + seg) * 32 + lane] = make_float2(sr, si);
}

// ---------------------------------------------------------------------------
// Stage 1b: full pass. One wave per (b,h,seg) for seg=0..3; composes the true
// initial state Init(seg) = sum_j w^(LSEG*(seg-1-j)) * Sloc(j), then scans its
// segment producing y = Re(sum_n C2_n s_n) + D*u, exact GELU.
// Mode partials -> LDS 32x32 transpose (pad 36 -> distinct banks).
// ---------------------------------------------------------------------------
__global__ __launch_bounds__(256) void s4d_scan_kernel(
    const float* __restrict__ u,
    const float* __restrict__ log_dt,
    const float* __restrict__ log_A_real,
    const float* __restrict__ A_imag,
    const float* __restrict__ C_real,
    const float* __restrict__ C_imag,
    const float* __restrict__ Dvec,
    const float2* __restrict__ Sloc,
    float* __restrict__ yact)
{
    __shared__ float pbuf[8][32][36];
    const int lane = threadIdx.x & 31;
    const int w    = threadIdx.x >> 5;
    const int wg   = blockIdx.x * 8 + w;   // [0, B*H*4)
    const int seg  = wg & 3;
    const int pair = wg >> 2;              // b*H + h
    const int h    = pair & (H_ - 1);

    const ModeK mk = mode_setup(h, lane, log_dt, log_A_real, A_imag);

    // C2 = 2*C*expm1(dtA)/A
    const int   hn = h * N2_ + lane;
    const float Ar = -expf(log_A_real[hn]);
    const float Ai = A_imag[hn];
    const float er = mk.wr - 1.0f;
    const float ei = mk.wi;
    const float Cr = C_real[hn];
    const float Ci = C_imag[hn];
    const float nr = Cr * er - Ci * ei;
    const float ni = Cr * ei + Ci * er;
    const float inv = 1.0f / (Ar * Ar + Ai * Ai);
    const float c2r = 2.0f * (nr * Ar + ni * Ai) * inv;
    const float c2i = 2.0f * (ni * Ar - nr * Ai) * inv;
    const float dD  = Dvec[h];

    // Compose initial state from previous segments' local end-states
    float sr = 0.0f, si = 0.0f;
    for (int j = 0; j < seg; ++j) {
        const float2 s = Sloc[(pair * 3 + j) * 32 + lane];
        const float  m = (float)(LSEG * (seg - 1 - j));
        const float  pe = expf(m * mk.dAr);
        const float  pr = pe * cosf(m * mk.dAi);
        const float  pi = pe * sinf(m * mk.dAi);
        sr += pr * s.x - pi * s.y;
        si += pr * s.y + pi * s.x;
    }

    const int base = pair * L_ + seg * LSEG;
    for (int lc = 0; lc < LSEG; lc += 32) {
        const float uvec = u[base + lc + lane];
#pragma unroll
        for (int j = 0; j < 32; ++j) {
            const float uv  = __shfl(uvec, j, 32);
            const float t_r = fmaf(mk.wr, sr, fmaf(-mk.wi, si, uv));
            const float t_i = fmaf(mk.wr, si, mk.wi * sr);
            sr = t_r; si = t_i;
            pbuf[w][j][lane] = fmaf(c2r, sr, -(c2i * si));
        }
        const float4* row = (const float4*)(&pbuf[w][lane][0]);   // 144B stride
        float y = 0.0f;
#pragma unroll
        for (int q = 0; q < 8; ++q) {
            const float4 v = row[q];
            y += (v.x + v.y) + (v.z + v.w);
        }
        y = fmaf(dD, uvec, y);
        const float g = 0.5f * y * (1.0f + erff(y * 0.70710678118654752f));
        yact[base + lc + lane] = g;
    }
}

// ---------------------------------------------------------------------------
// Stage 2: z = W(512x256) @ yact(256xL) via V_WMMA_F32_16X16X4_F32,
// fused bias + GLU + residual + channel LayerNorm. 512x64 tile per block.
// Wave w owns a-rows [32w,32w+32) AND matching g-rows (+256): GLU in regs.
// ---------------------------------------------------------------------------
__global__ __launch_bounds__(256) void gemm_glu_ln_kernel(
    const float* __restrict__ u,
    const float* __restrict__ yact,
    const float* __restrict__ W,        // (512,256) row-major
    const float* __restrict__ bconv,
    const float* __restrict__ gamma,
    const float* __restrict__ beta,
    float* __restrict__ out)
{
    __shared__ float Ys[H_ * S2];       // 256 x 64, stride 72
    __shared__ float redS[8][NCOL];
    __shared__ float redQ[8][NCOL];
    __shared__ float meanS[NCOL];
    __shared__ float rstdS[NCOL];

    const int tid  = threadIdx.x;
    const int lane = tid & 31;
    const int w    = tid >> 5;
    const int half = lane >> 4;
    const int l15  = lane & 15;
    const int b    = blockIdx.x >> 7;               // L_/NCOL == 128 tiles/batch
    const int l0   = (blockIdx.x & 127) * NCOL;

    // Cooperative tile fill: 256x64, coalesced (lane -> consecutive columns)
    {
        const int n  = tid & 63;
        const int k0 = tid >> 6;
#pragma unroll 4
        for (int rep = 0; rep < 64; ++rep) {
            const int k = rep * 4 + k0;
            const float* gp = &yact[(b * H_ + k) * L_ + l0 + n];
            float* lp = &Ys[k * S2 + n];
#if HAVE_ASYNC_LDS
            async_copy_f32(gp, lp);
#else
            *lp = *gp;
#endif
        }
#if HAVE_ASYNC_LDS
        __builtin_amdgcn_s_wait_asynccnt(0);
#endif
    }
    __syncthreads();

    v8f acc[4][4];
#pragma unroll
    for (int mt = 0; mt < 4; ++mt)
#pragma unroll
        for (int nt = 0; nt < 4; ++nt)
            acc[mt][nt] = (v8f){0.f,0.f,0.f,0.f,0.f,0.f,0.f,0.f};

    const int mrow0 = 32 * w;

    for (int k = 0; k < H_; k += 4) {
        if ((k & 63) == 0) {             // prefetch W one stripe ahead (L2->L0)
#pragma unroll
            for (int mt = 0; mt < 4; ++mt) {
                const int row = mrow0 + ((mt & 1) ? 16 : 0) + ((mt & 2) ? 256 : 0) + l15;
                __builtin_prefetch(&W[row * H_ + k + 64], 0, 1);
            }
        }
        v2f bf[4];
#pragma unroll
        for (int nt = 0; nt < 4; ++nt) {
            const int col = nt * 16 + l15;
            const int kr  = k + half * 2;
            v2f t;
            t.x = Ys[kr * S2 + col];
            t.y = Ys[(kr + 1) * S2 + col];
            bf[nt] = t;
        }
#pragma unroll
        for (int mt = 0; mt < 4; ++mt) {
            const int row = mrow0 + ((mt & 1) ? 16 : 0) + ((mt & 2) ? 256 : 0) + l15;
            const v2f a = *(const v2f*)(&W[row * H_ + k + half * 2]);
#pragma unroll
            for (int nt = 0; nt < 4; ++nt)
                acc[mt][nt] = __builtin_amdgcn_wmma_f32_16x16x4_f32(
                    false, a, false, bf[nt], (short)0, acc[mt][nt], false, false);
        }
    }

    // Epilogue: bias + GLU + residual; LN stats per column
    float ps[4] = {0.f,0.f,0.f,0.f}, pq[4] = {0.f,0.f,0.f,0.f};
#pragma unroll
    for (int mt = 0; mt < 2; ++mt) {
#pragma unroll
        for (int v = 0; v < 8; ++v) {
            const int c  = mrow0 + (mt ? 16 : 0) + v + half * 8;
            const float ba = bconv[c];
            const float bg = bconv[c + 256];
#pragma unroll
            for (int nt = 0; nt < 4; ++nt) {
                const int n = nt * 16 + l15;
                const float av = acc[mt][nt][v] + ba;
                const float gv = acc[mt + 2][nt][v] + bg;
                const float x  = av * (1.0f / (1.0f + expf(-gv)));
                const float r  = u[(b * H_ + c) * L_ + l0 + n] + x;
                acc[mt][nt][v] = r;
                ps[nt] += r;
                pq[nt]  = fmaf(r, r, pq[nt]);
            }
        }
    }
#pragma unroll
    for (int nt = 0; nt < 4; ++nt) {     // lanes l and l^16 share column N=lane%16
        ps[nt] += __shfl_xor(ps[nt], 16, 32);
        pq[nt] += __shfl_xor(pq[nt], 16, 32);
    }
    if (lane < 16) {
#pragma unroll
        for (int nt = 0; nt < 4; ++nt) {
            redS[w][nt * 16 + l15] = ps[nt];
            redQ[w][nt * 16 + l15] = pq[nt];
        }
    }
    __syncthreads();
    if (tid < NCOL) {
        float s = 0.f, q = 0.f;
#pragma unroll
        for (int ww = 0; ww < 8; ++ww) { s += redS[ww][tid]; q += redQ[ww][tid]; }
        const float mean = s * (1.0f / 256.0f);
        const float var  = q * (1.0f / 256.0f) - mean * mean;
        meanS[tid] = mean;
        rstdS[tid] = rsqrtf(var + LN_EPS);
    }
    __syncthreads();

#pragma unroll
    for (int mt = 0; mt < 2; ++mt) {
#pragma unroll
        for (int v = 0; v < 8; ++v) {
            const int c  = mrow0 + (mt ? 16 : 0) + v + half * 8;
            const float gm = gamma[c];
            const float bt = beta[c];
#pragma unroll
            for (int nt = 0; nt < 4; ++nt) {
                const int n = nt * 16 + l15;
                const float r = acc[mt][nt][v];
                out[(b * H_ + c) * L_ + l0 + n] = (r - meanS[n]) * rstdS[n] * gm + bt;
            }
        }
    }
}

extern "C" void kernel_launch(void* const* d_in, const int* in_sizes, int n_in,
                              void* d_out, int out_size, void* d_ws, size_t ws_size,
                              hipStream_t stream) {
    const float* u          = (const float*)d_in[0];
    const float* log_dt     = (const float*)d_in[1];
    const float* log_A_real = (const float*)d_in[2];
    const float* A_imag     = (const float*)d_in[3];
    const float* C_real     = (const float*)d_in[4];
    const float* C_imag     = (const float*)d_in[5];
    const float* Dvec       = (const float*)d_in[6];
    const float* W          = (const float*)d_in[7];
    const float* bconv      = (const float*)d_in[8];
    const float* gamma      = (const float*)d_in[9];
    const float* beta       = (const float*)d_in[10];

    float2* Sloc = (float2*)d_ws;                         // 1.5 MiB
    float*  yact = (float*)((char*)d_ws + (2u << 20));    // 64 MiB @ +2MiB
    float*  out  = (float*)d_out;

    s4d_state_kernel<<<(B_ * H_ * 3) / 8, 256, 0, stream>>>(
        u, log_dt, log_A_real, A_imag, Sloc);
    s4d_scan_kernel<<<(B_ * H_ * NSEG) / 8, 256, 0, stream>>>(
        u, log_dt, log_A_real, A_imag, C_real, C_imag, Dvec, Sloc, yact);
    gemm_glu_ln_kernel<<<B_ * (L_ / NCOL), 256, 0, stream>>>(
        u, yact, W, bconv, gamma, beta, out);
}